// LightpathGNN_45191645888550
// MI455X (gfx1250) — compile-verified
//
#include <hip/hip_runtime.h>
#include <hip/hip_bf16.h>

typedef __attribute__((ext_vector_type(2))) float v2f;
typedef __attribute__((ext_vector_type(8))) float v8f;

#define HC 128   // H*C
#define NH 4     // heads
#define IN_F 16  // input features

__device__ __forceinline__ float leaky2(float v) { return v >= 0.f ? v : 0.2f * v; }

__device__ __forceinline__ void atomicMaxF32(float* addr, float val) {
    // monotone float max via sign-split integer atomics (works for mixed signs)
    if (val >= 0.0f) atomicMax((int*)addr, __float_as_int(val));
    else             atomicMin((unsigned int*)addr, __float_as_uint(val));
}

// ---------------------------------------------------------------------------
// K1: h = x @ W_lin   via V_WMMA_F32_16X16X4_F32  (16-node tiles, K=16, N=128)
// A 16x4 f32 layout: lane = (k>=2)*16 + m, vgpr = k%2 (per 4-wide K step)
// C/D 16x16 f32   : lane = (m/8)*16 + n, vgpr = m%8
// ---------------------------------------------------------------------------
__global__ void k_node_linear(const float* __restrict__ x, const float* __restrict__ W,
                              float* __restrict__ h, int N) {
    int wave = (blockIdx.x * blockDim.x + threadIdx.x) >> 5;
    int lane = threadIdx.x & 31;
    int tiles = (N + 15) >> 4;
    if (wave >= tiles) return;
    int half = lane >> 4;        // K-half selector for A/B, M-half for D
    int mrow = lane & 15;        // A row
    int ncol = lane & 15;        // B/D column
    int node = wave * 16 + mrow; // N = 100000 is a multiple of 16

    v2f afr[4];
#pragma unroll
    for (int s = 0; s < 4; s++) {
        int k0 = s * 4 + half * 2;
        const float* xr = x + (size_t)node * IN_F;
        afr[s].x = xr[k0];
        afr[s].y = xr[k0 + 1];
    }
#pragma unroll
    for (int t = 0; t < 8; t++) {          // 8 column tiles of 16 -> 128 cols
        v8f acc = {};
#pragma unroll
        for (int s = 0; s < 4; s++) {      // K = 16 in steps of 4
            int k0 = s * 4 + half * 2;
            v2f b;
            b.x = W[(size_t)k0 * HC + t * 16 + ncol];
            b.y = W[(size_t)(k0 + 1) * HC + t * 16 + ncol];
            acc = __builtin_amdgcn_wmma_f32_16x16x4_f32(
                false, afr[s], false, b, (short)0, acc, false, false);
        }
#pragma unroll
        for (int r = 0; r < 8; r++) {
            int row = wave * 16 + half * 8 + r;
            h[(size_t)row * HC + t * 16 + ncol] = acc[r];
        }
    }
}

// ---------------------------------------------------------------------------
// K2: per-node attention scores; seed segment-max with the self-loop value;
//     zero the aggregation buffer and denominators.  1 node / 128-thread block
// ---------------------------------------------------------------------------
__global__ void k_scores(const float* __restrict__ h, const float* __restrict__ att_src,
                         const float* __restrict__ att_dst, float* __restrict__ a_src,
                         float* __restrict__ a_dst, float* __restrict__ mmax,
                         float* __restrict__ denom, float* __restrict__ aggr, int N) {
    int n = blockIdx.x;
    if (n >= N) return;
    int c = threadIdx.x;          // 0..127
    int head = c >> 5;            // wave id == head (wave32)
    float hv = h[(size_t)n * HC + c];
    float vs = hv * att_src[c];
    float vd = hv * att_dst[c];
#pragma unroll
    for (int off = 16; off; off >>= 1) {
        vs += __shfl_xor(vs, off, 32);
        vd += __shfl_xor(vd, off, 32);
    }
    aggr[(size_t)n * HC + c] = 0.f;
    if ((c & 31) == 0) {
        a_src[n * NH + head] = vs;
        a_dst[n * NH + head] = vd;
        mmax[n * NH + head]  = leaky2(vs + vd);   // self-loop seeds the max
        denom[n * NH + head] = 0.f;
    }
}

// ---------------------------------------------------------------------------
// K3: edge pass 1 — segment max (atomic float max), one edge per thread
// ---------------------------------------------------------------------------
__global__ void k_edge_max(const int* __restrict__ src, const int* __restrict__ dst,
                           const float* __restrict__ a_src, const float* __restrict__ a_dst,
                           float* __restrict__ mmax, int E) {
    int e = blockIdx.x * blockDim.x + threadIdx.x;
    if (e >= E) return;
    int s = src[e], d = dst[e];
#pragma unroll
    for (int hh = 0; hh < NH; hh++) {
        float v = leaky2(a_src[s * NH + hh] + a_dst[d * NH + hh]);
        atomicMaxF32(&mmax[d * NH + hh], v);
    }
}

// ---------------------------------------------------------------------------
// K4: edge pass 2 — unnormalized softmax-weighted aggregation.
//     One edge per wave32: lane handles 4 channels (float4, coalesced 512B).
//     Normalization deferred to K5 (denominator constant per (dst,head)).
// ---------------------------------------------------------------------------
__global__ void k_edge_aggr(const int* __restrict__ src, const int* __restrict__ dst,
                            const float* __restrict__ a_src, const float* __restrict__ a_dst,
                            const float* __restrict__ mmax, const float* __restrict__ h,
                            float* __restrict__ denom, float* __restrict__ aggr, int E) {
    int wid  = (blockIdx.x * blockDim.x + threadIdx.x) >> 5;
    int lane = threadIdx.x & 31;
    if (wid >= E) return;
    int s = src[wid], d = dst[wid];
    int head = lane >> 3;                  // channels lane*4..lane*4+3 -> head
    float ev = __expf(leaky2(a_src[s * NH + head] + a_dst[d * NH + head])
                      - mmax[d * NH + head]);
    if (lane < NH) {
        float v2 = leaky2(a_src[s * NH + lane] + a_dst[d * NH + lane]);
        atomicAdd(&denom[d * NH + lane], __expf(v2 - mmax[d * NH + lane]));
    }
    const float4 hv = ((const float4*)(h + (size_t)s * HC))[lane];
    int base = d * HC + lane * 4;
    atomicAdd(&aggr[base + 0], hv.x * ev);
    atomicAdd(&aggr[base + 1], hv.y * ev);
    atomicAdd(&aggr[base + 2], hv.z * ev);
    atomicAdd(&aggr[base + 3], hv.w * ev);
}

// K4b: zero the BatchNorm accumulators (must happen every launch)
__global__ void k_zero(float* __restrict__ p, int n) {
    int i = blockIdx.x * blockDim.x + threadIdx.x;
    if (i < n) p[i] = 0.f;
}

// ---------------------------------------------------------------------------
// K5: add self-loop message, normalize, + bias  (in place in aggr);
//     accumulate BN partial sums (one atomicAdd per thread per block).
// ---------------------------------------------------------------------------
__global__ void k_finalize(const float* __restrict__ h, const float* __restrict__ a_src,
                           const float* __restrict__ a_dst, const float* __restrict__ mmax,
                           const float* __restrict__ denom, const float* __restrict__ bias,
                           float* __restrict__ aggr, float* __restrict__ bnsum,
                           float* __restrict__ bnsq, int N) {
    int c = threadIdx.x;          // 0..127
    int head = c >> 5;
    float lsum = 0.f, lsq = 0.f;
    for (int n = blockIdx.x; n < N; n += gridDim.x) {
        float sv    = leaky2(a_src[n * NH + head] + a_dst[n * NH + head]);
        float eself = __expf(sv - mmax[n * NH + head]);   // <= 1, stable
        float dt    = denom[n * NH + head] + eself;       // > 0 always
        float val = (aggr[(size_t)n * HC + c] + h[(size_t)n * HC + c] * eself) / dt
                    + bias[c];
        aggr[(size_t)n * HC + c] = val;
        lsum += val;
        lsq  += val * val;
    }
    atomicAdd(&bnsum[c], lsum);
    atomicAdd(&bnsq[c], lsq);
}

// K6: fold BN stats into affine coefficients: y = relu(v*coefA + coefB)
__global__ void k_bnstats(const float* __restrict__ bnsum, const float* __restrict__ bnsq,
                          const float* __restrict__ gamma, const float* __restrict__ beta,
                          float* __restrict__ coefA, float* __restrict__ coefB, int N) {
    int c = threadIdx.x;
    if (c >= HC) return;
    float invN = 1.f / (float)N;
    float mean = bnsum[c] * invN;
    float var  = bnsq[c] * invN - mean * mean;   // biased var (jnp.var default)
    float a = gamma[c] * rsqrtf(var + 1e-5f);
    coefA[c] = a;
    coefB[c] = beta[c] - mean * a;
}

// K7: gather LUT nodes, apply BN + ReLU -> emb [NL,128]
__global__ void k_gather_bn(const float* __restrict__ aggr, const int* __restrict__ lut,
                            const float* __restrict__ coefA, const float* __restrict__ coefB,
                            float* __restrict__ emb, int NL) {
    int gid = blockIdx.x * blockDim.x + threadIdx.x;
    if (gid >= NL * HC) return;
    int i = gid >> 7, c = gid & (HC - 1);
    int n = lut[i];
    float v = aggr[(size_t)n * HC + c] * coefA[c] + coefB[c];
    emb[gid] = v > 0.f ? v : 0.f;
}

// ---------------------------------------------------------------------------
// K8: z = leaky(emb @ W1 + b1, 0.01)  — WMMA f32 16x16x4, K=128, Ncols=32
// ---------------------------------------------------------------------------
__global__ void k_mlp1(const float* __restrict__ emb, const float* __restrict__ W1,
                       const float* __restrict__ b1, float* __restrict__ z, int M) {
    int wave = (blockIdx.x * blockDim.x + threadIdx.x) >> 5;
    int lane = threadIdx.x & 31;
    int mtiles = M >> 4;                 // M = 1024 -> 64
    int mt = wave >> 1, nt = wave & 1;   // 2 column tiles of 16 -> 32 cols
    if (mt >= mtiles) return;
    int half = lane >> 4;
    int mrow = lane & 15;
    int ncol = lane & 15;
    const float* arow = emb + (size_t)(mt * 16 + mrow) * HC;
    v8f acc = {};
    for (int s = 0; s < 32; s++) {       // K = 128 in steps of 4
        int k0 = s * 4 + half * 2;
        v2f a, b;
        a.x = arow[k0];
        a.y = arow[k0 + 1];
        b.x = W1[(size_t)k0 * 32 + nt * 16 + ncol];
        b.y = W1[(size_t)(k0 + 1) * 32 + nt * 16 + ncol];
        acc = __builtin_amdgcn_wmma_f32_16x16x4_f32(
            false, a, false, b, (short)0, acc, false, false);
    }
    int col = nt * 16 + ncol;
    float bb = b1[col];
#pragma unroll
    for (int r = 0; r < 8; r++) {
        int row = mt * 16 + half * 8 + r;
        float v = acc[r] + bb;
        z[(size_t)row * 32 + col] = v >= 0.f ? v : 0.01f * v;
    }
}

// K9: out = z @ W2 + b2   (32-wide dot per row)
__global__ void k_mlp2(const float* __restrict__ z, const float* __restrict__ W2,
                       const float* __restrict__ b2, float* __restrict__ out, int M) {
    int i = blockIdx.x * blockDim.x + threadIdx.x;
    if (i >= M) return;
    float acc = b2[0];
#pragma unroll
    for (int j = 0; j < 32; j++) acc += z[i * 32 + j] * W2[j];
    out[i] = acc;
}

// ---------------------------------------------------------------------------
extern "C" void kernel_launch(void* const* d_in, const int* in_sizes, int n_in,
                              void* d_out, int out_size, void* d_ws, size_t ws_size,
                              hipStream_t stream) {
    const float* x       = (const float*)d_in[0];
    const int*   edge    = (const int*)d_in[1];   // int32 (JAX x64 disabled by default)
    const int*   lut     = (const int*)d_in[2];
    const float* W_lin   = (const float*)d_in[3];
    const float* att_src = (const float*)d_in[4];
    const float* att_dst = (const float*)d_in[5];
    const float* bias_g  = (const float*)d_in[6];
    const float* gamma   = (const float*)d_in[7];
    const float* beta    = (const float*)d_in[8];
    const float* W1      = (const float*)d_in[9];
    const float* b1      = (const float*)d_in[10];
    const float* W2      = (const float*)d_in[11];
    const float* b2      = (const float*)d_in[12];
    float* out = (float*)d_out;

    const int N  = in_sizes[0] / IN_F;     // 100000
    const int E  = in_sizes[1] / 2;        // 1600000
    const int NL = in_sizes[2];            // 1024
    const int* src = edge;
    const int* dst = edge + E;

    // workspace layout (floats)
    float* ws     = (float*)d_ws;
    size_t N128   = (size_t)N * HC;
    size_t N4     = (size_t)N * NH;
    float* h      = ws;                 // [N,128]
    float* aggr   = h + N128;           // [N,128]  (becomes post-GAT/BN input)
    float* a_src  = aggr + N128;        // [N,4]
    float* a_dst  = a_src + N4;         // [N,4]
    float* mmax   = a_dst + N4;         // [N,4]
    float* denom  = mmax + N4;          // [N,4]
    float* bnsum  = denom + N4;         // [128]
    float* bnsq   = bnsum + HC;         // [128]
    float* coefA  = bnsq + HC;          // [128]
    float* coefB  = coefA + HC;         // [128]
    float* emb    = coefB + HC;         // [NL,128]
    float* z      = emb + (size_t)NL * HC; // [NL,32]

    // K1: node linear (WMMA).  8 waves/block, 1 tile of 16 nodes per wave.
    {
        int tiles = (N + 15) / 16;
        int blocks = (tiles + 7) / 8;
        k_node_linear<<<blocks, 256, 0, stream>>>(x, W_lin, h, N);
    }
    // K2: scores + init
    k_scores<<<N, HC, 0, stream>>>(h, att_src, att_dst, a_src, a_dst, mmax, denom, aggr, N);
    // K3: edge segment-max
    k_edge_max<<<(E + 255) / 256, 256, 0, stream>>>(src, dst, a_src, a_dst, mmax, E);
    // K4: edge aggregation (1 edge / wave32, 8 edges / block)
    k_edge_aggr<<<(E + 7) / 8, 256, 0, stream>>>(src, dst, a_src, a_dst, mmax, h, denom, aggr, E);
    // K4b: zero BN accumulators
    k_zero<<<1, 256, 0, stream>>>(bnsum, 2 * HC);
    // K5: finalize GAT output + BN partial sums
    k_finalize<<<1024, HC, 0, stream>>>(h, a_src, a_dst, mmax, denom, bias_g, aggr, bnsum, bnsq, N);
    // K6: BN coefficients
    k_bnstats<<<1, HC, 0, stream>>>(bnsum, bnsq, gamma, beta, coefA, coefB, N);
    // K7: LUT gather + BN + ReLU
    k_gather_bn<<<(NL * HC + 255) / 256, 256, 0, stream>>>(aggr, lut, coefA, coefB, emb, NL);
    // K8: MLP layer 1 (WMMA): 64 mtiles x 2 ntiles = 128 waves
    {
        int waves = (NL / 16) * 2;
        int blocks = (waves + 7) / 8;
        k_mlp1<<<blocks, 256, 0, stream>>>(emb, W1, b1, z, NL);
    }
    // K9: MLP layer 2
    k_mlp2<<<(NL + 255) / 256, 256, 0, stream>>>(z, W2, b2, out, NL);
}